// GCNLayer_30416958390405
// MI455X (gfx1250) — compile-verified
//
#include <hip/hip_runtime.h>

typedef __attribute__((ext_vector_type(2))) float v2f;
typedef __attribute__((ext_vector_type(8))) float v8f;

// Problem constants (from reference)
constexpr int Bc   = 8;
constexpr int Nc   = 50000;
constexpr int DINc = 16;
constexpr int Ec   = 800000;
constexpr int Mc   = 25000;
constexpr int BCc  = Bc * DINc;          // 128 aggregated feature columns
constexpr int TILES = (Mc + 15) / 16;    // 1563 centroid tiles

__device__ __forceinline__ v8f wmma4(v2f a, v2f b, v8f c) {
  // D = A(16x4,f32) * B(4x16,f32) + C(16x16,f32)  -> v_wmma_f32_16x16x4_f32
  return __builtin_amdgcn_wmma_f32_16x16x4_f32(false, a, false, b, (short)0, c,
                                               false, false);
}

// ---- Kernel 1: zero agg (N*128 f32) and flags (N bytes, as N/4 words) ----
__global__ void zero_ws_kernel(float* agg, unsigned int* flag_words) {
  const long long aggWords = (long long)Nc * BCc;
  const long long flagWords = Nc / 4;
  long long i = (long long)blockIdx.x * blockDim.x + threadIdx.x;
  if (i < aggWords) {
    agg[i] = 0.0f;
  } else if (i < aggWords + flagWords) {
    flag_words[i - aggWords] = 0u;
  }
}

// ---- Kernel 2: mark nodes that are centroids ----
__global__ void mark_centroids_kernel(const int* __restrict__ centroids,
                                      unsigned char* __restrict__ flags) {
  int i = blockIdx.x * blockDim.x + threadIdx.x;
  if (i < Mc) flags[centroids[i]] = 1;
}

// ---- Kernel 3: edge scatter; one 128-thread block per edge ----
__global__ __launch_bounds__(128) void edge_scatter_kernel(
    const int* __restrict__ rows, const int* __restrict__ cols,
    const float* __restrict__ vals, const float* __restrict__ x,
    const unsigned char* __restrict__ flags, float* __restrict__ agg) {
  int e = blockIdx.x;
  int r = rows[e];
  if (!flags[r]) return;                 // row never read by output -> skip
  int col = cols[e];
  float v = vals[e];
  int t = threadIdx.x;                   // 0..127 == b*16 + c
  int b = t >> 4;
  int c = t & 15;
  float xv = x[((long long)b * Nc + col) * DINc + c];
  atomicAdd(&agg[(long long)r * BCc + t], v * xv);
}

// ---- Kernel 4: gather centroids + dual 16->32 GEMM via WMMA + bias + ReLU ----
__global__ __launch_bounds__(256) void gemm_tiles_kernel(
    const float* __restrict__ x, const float* __restrict__ agg,
    const int* __restrict__ centroids,
    const float* __restrict__ W_lin, const float* __restrict__ b_lin,
    const float* __restrict__ W_eye, const float* __restrict__ b_eye,
    float* __restrict__ out) {
  int wave = threadIdx.x >> 5;
  int lane = threadIdx.x & 31;
  int tile = blockIdx.x * (blockDim.x >> 5) + wave;
  if (tile >= Bc * TILES) return;        // uniform per wave: EXEC stays full

  int b  = tile / TILES;
  int t  = tile % TILES;
  int m0 = t * 16;
  int row = lane & 15;                   // A: M-row | B: N-col | D: N-col
  int hi  = lane >> 4;                   // K-half selector for A/B fragments
  int m = m0 + row;
  if (m >= Mc) m = Mc - 1;               // clamp loads for partial last tile
  int node = centroids[m];

  // A fragments (16x4 f32 layout): lane reads K = 4k + 2*hi, +1 as float2
  const v2f* xr = (const v2f*)(x + ((long long)b * Nc + node) * DINc);
  const v2f* ar = (const v2f*)(agg + (long long)node * BCc + b * DINc);
  v2f aE[4], aL[4];
#pragma unroll
  for (int k = 0; k < 4; ++k) {
    aE[k] = xr[k * 2 + hi];
    aL[k] = ar[k * 2 + hi];
  }

  // B fragments (4x16 f32): lane reads W[otile*16 + row][4k + 2*hi .. +1]
  v2f bL[2][4], bE[2][4];
#pragma unroll
  for (int ot = 0; ot < 2; ++ot) {
    const v2f* wl = (const v2f*)(W_lin + (ot * 16 + row) * DINc);
    const v2f* we = (const v2f*)(W_eye + (ot * 16 + row) * DINc);
#pragma unroll
    for (int k = 0; k < 4; ++k) {
      bL[ot][k] = wl[k * 2 + hi];
      bE[ot][k] = we[k * 2 + hi];
    }
  }

  // C init = bias (depends only on output channel = D's N-col = row)
  float bias[4] = { b_lin[row], b_lin[16 + row], b_eye[row], b_eye[16 + row] };
  v8f acc[4];
#pragma unroll
  for (int i = 0; i < 4; ++i) {
    v8f c;
#pragma unroll
    for (int r = 0; r < 8; ++r) c[r] = bias[i];
    acc[i] = c;
  }

  // K=16 as 4 chained K=4 WMMAs; 4 independent D tiles interleave nicely
#pragma unroll
  for (int k = 0; k < 4; ++k) {
    acc[0] = wmma4(aL[k], bL[0][k], acc[0]);   // lin, channels  0..15
    acc[1] = wmma4(aL[k], bL[1][k], acc[1]);   // lin, channels 16..31
    acc[2] = wmma4(aE[k], bE[0][k], acc[2]);   // eye, channels 32..47
    acc[3] = wmma4(aE[k], bE[1][k], acc[3]);   // eye, channels 48..63
  }

  // Store with ReLU. D layout: lane holds N-col = row, vgpr r -> M = r + 8*hi
#pragma unroll
  for (int i = 0; i < 4; ++i) {
    int obase = i * 16;
#pragma unroll
    for (int r = 0; r < 8; ++r) {
      int mr = m0 + r + hi * 8;
      if (mr < Mc) {
        float v = acc[i][r];
        out[((long long)b * Mc + mr) * 64 + obase + row] = v > 0.0f ? v : 0.0f;
      }
    }
  }
}

extern "C" void kernel_launch(void* const* d_in, const int* in_sizes, int n_in,
                              void* d_out, int out_size, void* d_ws, size_t ws_size,
                              hipStream_t stream) {
  const float* x         = (const float*)d_in[0];
  const int*   adj_rows  = (const int*)d_in[1];
  const int*   adj_cols  = (const int*)d_in[2];
  const float* adj_vals  = (const float*)d_in[3];
  const int*   centroids = (const int*)d_in[4];
  const float* W_lin     = (const float*)d_in[5];
  const float* b_lin     = (const float*)d_in[6];
  const float* W_eye     = (const float*)d_in[7];
  const float* b_eye     = (const float*)d_in[8];
  float* out = (float*)d_out;

  float* agg = (float*)d_ws;                                   // N*128 f32
  unsigned char* flags = (unsigned char*)d_ws + (size_t)Nc * BCc * 4;  // N bytes

  // 1) zero workspace (deterministic each call)
  {
    long long words = (long long)Nc * BCc + Nc / 4;
    int blocks = (int)((words + 255) / 256);
    zero_ws_kernel<<<blocks, 256, 0, stream>>>(agg, (unsigned int*)flags);
  }
  // 2) mark centroid nodes
  mark_centroids_kernel<<<(Mc + 255) / 256, 256, 0, stream>>>(centroids, flags);
  // 3) sparse scatter (skips rows no output reads)
  edge_scatter_kernel<<<Ec, 128, 0, stream>>>(adj_rows, adj_cols, adj_vals, x,
                                              flags, agg);
  // 4) fused gather + WMMA GEMMs + bias + ReLU
  {
    int tiles = Bc * TILES;
    int blocks = (tiles + 7) / 8;        // 8 waves (tiles) per 256-thread block
    gemm_tiles_kernel<<<blocks, 256, 0, stream>>>(x, agg, centroids, W_lin,
                                                  b_lin, W_eye, b_eye, out);
  }
}